// CoopProxyLoss_79731772883630
// MI455X (gfx1250) — compile-verified
//
#include <hip/hip_runtime.h>
#include <hip/hip_bf16.h>

// ---------------------------------------------------------------------------
// CDNA5 (gfx1250) implementation of CoopProxyLoss using V_WMMA_F32_16X16X4_F32
// for all dot-product stages (fp32 matrix path, wave32 tiles), with
// GLOBAL_LOAD_ASYNC_TO_LDS staging for the embedding GEMM and register-resident
// A-fragments in the streaming distance/softmax heads.
// ---------------------------------------------------------------------------

typedef float v2f __attribute__((ext_vector_type(2)));
typedef float v8f __attribute__((ext_vector_type(8)));

#define N_   4096
#define F_   1024
#define D_   128
#define P_   8192
#define C_   64
#define A_   4
#define V_   256

#define NEG_BIG (-3.0e38f)

__device__ __forceinline__ v8f wmma_f32(v2f a, v2f b, v8f c) {
    // D = A(16x4 f32) * B(4x16 f32) + C(16x16 f32)
    return __builtin_amdgcn_wmma_f32_16x16x4_f32(
        /*neg_a=*/false, a, /*neg_b=*/false, b,
        /*c_mod=*/(short)0, c, /*reuse_a=*/false, /*reuse_b=*/false);
}

// ---------------- workspace layout (float offsets) ----------------
#define O_EMB     0                        // N*D       = 524288
#define O_XX      524288                   // N         = 4096
#define O_ZZP     528384                   // P         = 8192
#define O_SC      536576                   // C*D       = 8192 (sums -> centers)
#define O_SCCNT   544768                   // C         = 64
#define O_CATCNT  544832                   // C         = 64
#define O_ACC     544896                   // 4 accumulators (ip, cp, mm, reg)
#define O_ZZSC    544900                   // C         = 64
#define O_W2      544964                   // A*V*D     = 131072
#define O_ZZW     676036                   // A*V       = 1024
#define O_MXX     677060                   // A*N       = 16384
#define ZERO_LEN  8324                     // sc + sccnt + catcnt + acc (contiguous)

// ---------------- init ----------------
__global__ void k_zero(float* p, int n) {
    int i = blockIdx.x * blockDim.x + threadIdx.x;
    if (i < n) p[i] = 0.0f;
}

// ---------------- emb = images @ W_model  (WMMA fp32) ----------------
// grid: N/16 blocks of 256 threads (8 waves). wave w owns col tile w of D=128.
// The 16x1024 A strip is staged once per block into LDS with
// GLOBAL_LOAD_ASYNC_TO_LDS_B128 (ASYNCcnt) and shared by all 8 waves.
__global__ void k_gemm_emb(const float* __restrict__ X,
                           const float* __restrict__ W,
                           float* __restrict__ E) {
    __shared__ float ldsX[16 * F_];        // 64 KB
    const int tid  = threadIdx.x;
    const int wave = tid >> 5;
    const int lane = tid & 31;
    const int r = lane & 15, half = lane >> 4, kb = half * 2;
    const int row0 = blockIdx.x * 16;
    const int col0 = wave * 16;

    // ---- async stage: 16 rows x 1024 f32 = 4096 x 16B chunks, 256 thr x 16 it
    {
        const unsigned long long gbase =
            (unsigned long long)(size_t)(X + (size_t)row0 * F_);
        const unsigned lbase = (unsigned)(size_t)ldsX;   // low 32b = LDS offset
#pragma unroll
        for (int i = 0; i < 16; ++i) {
            const unsigned idx = (unsigned)(i * 256 + tid);     // 16B chunk id
            unsigned long long ga = gbase + (unsigned long long)idx * 16ull;
            unsigned la = lbase + idx * 16u;
            asm volatile("global_load_async_to_lds_b128 %0, %1, off"
                         :: "v"(la), "v"(ga) : "memory");
        }
        asm volatile("s_wait_asynccnt 0x0" ::: "memory");
    }
    __syncthreads();

    v8f acc = {0.f, 0.f, 0.f, 0.f, 0.f, 0.f, 0.f, 0.f};
    const float* Xr = ldsX + r * F_;       // LDS-resident A row
    for (int k = 0; k < F_; k += 4) {
        v2f a, b;
        a.x = Xr[k + kb];
        a.y = Xr[k + kb + 1];
        b.x = W[(size_t)(k + kb) * D_ + col0 + r];
        b.y = W[(size_t)(k + kb + 1) * D_ + col0 + r];
        acc = wmma_f32(a, b, acc);
    }
#pragma unroll
    for (int v = 0; v < 8; ++v)
        E[(size_t)(row0 + half * 8 + v) * D_ + col0 + r] = acc[v];
}

// ---------------- per-row stats of emb: xx, mxx[a], reg_e, cat counts -------
// one wave per row; 8 waves / block
__global__ void k_row_stats(const float* __restrict__ E,
                            const float* __restrict__ masks,
                            const int* __restrict__ cats,
                            float* __restrict__ xx, float* __restrict__ mxx,
                            float* __restrict__ catcnt, float* __restrict__ acc) {
    const int row  = blockIdx.x * 8 + (threadIdx.x >> 5);
    const int lane = threadIdx.x & 31;
    float4 e = ((const float4*)(E + (size_t)row * D_))[lane];
    float s = e.x * e.x + e.y * e.y + e.z * e.z + e.w * e.w;
    float sm[A_];
#pragma unroll
    for (int a = 0; a < A_; ++a) {
        float4 m = ((const float4*)(masks + a * D_))[lane];
        float x0 = e.x * m.x, x1 = e.y * m.y, x2 = e.z * m.z, x3 = e.w * m.w;
        sm[a] = x0 * x0 + x1 * x1 + x2 * x2 + x3 * x3;
    }
    for (int off = 16; off >= 1; off >>= 1) {
        s += __shfl_xor(s, off, 32);
#pragma unroll
        for (int a = 0; a < A_; ++a) sm[a] += __shfl_xor(sm[a], off, 32);
    }
    if (lane == 0) {
        xx[row] = s;
#pragma unroll
        for (int a = 0; a < A_; ++a) mxx[a * N_ + row] = sm[a];
        atomicAdd(&acc[3], sqrtf(s));
        atomicAdd(&catcnt[cats[row]], 1.0f);
    }
}

// ---------------- per-proxy stats: zz_p + segment sums ----------------------
__global__ void k_proxy_stats(const float* __restrict__ Z,
                              const int* __restrict__ pcats,
                              float* __restrict__ zz,
                              float* __restrict__ sc_sums,
                              float* __restrict__ sc_cnt) {
    const int p    = blockIdx.x * 8 + (threadIdx.x >> 5);
    const int lane = threadIdx.x & 31;
    float4 z = ((const float4*)(Z + (size_t)p * D_))[lane];
    float s  = z.x * z.x + z.y * z.y + z.z * z.z + z.w * z.w;
    for (int off = 16; off >= 1; off >>= 1) s += __shfl_xor(s, off, 32);
    const int c = pcats[p];
    float* dst = sc_sums + (size_t)c * D_ + lane * 4;
    atomicAdd(dst + 0, z.x);
    atomicAdd(dst + 1, z.y);
    atomicAdd(dst + 2, z.z);
    atomicAdd(dst + 3, z.w);
    if (lane == 0) {
        zz[p] = s;
        atomicAdd(&sc_cnt[c], 1.0f);
    }
}

// ---------------- supercenters = sums/counts; zz_sc ----------------
__global__ void k_centers(float* __restrict__ sc, const float* __restrict__ cnt,
                          float* __restrict__ zz_sc) {
    const int c    = blockIdx.x * 8 + (threadIdx.x >> 5);
    const int lane = threadIdx.x & 31;
    const float inv = 1.0f / cnt[c];
    float4* row = (float4*)(sc + (size_t)c * D_);
    float4 v = row[lane];
    v.x *= inv; v.y *= inv; v.z *= inv; v.w *= inv;
    row[lane] = v;
    float s = v.x * v.x + v.y * v.y + v.z * v.z + v.w * v.w;
    for (int off = 16; off >= 1; off >>= 1) s += __shfl_xor(s, off, 32);
    if (lane == 0) zz_sc[c] = s;
}

// ---------------- W2 = adj*mask^2 ; zzW = ||adj*mask||^2 ----------------
__global__ void k_mask_adj(const float* __restrict__ adj,
                           const float* __restrict__ masks,
                           float* __restrict__ W2, float* __restrict__ zzW) {
    const int idx  = blockIdx.x * 8 + (threadIdx.x >> 5);   // a*V + v
    const int a    = idx >> 8;
    const int lane = threadIdx.x & 31;
    float4 z = ((const float4*)(adj + (size_t)idx * D_))[lane];
    float4 m = ((const float4*)(masks + a * D_))[lane];
    float zm0 = z.x * m.x, zm1 = z.y * m.y, zm2 = z.z * m.z, zm3 = z.w * m.w;
    float4 w2; w2.x = zm0 * m.x; w2.y = zm1 * m.y; w2.z = zm2 * m.z; w2.w = zm3 * m.w;
    ((float4*)(W2 + (size_t)idx * D_))[lane] = w2;
    float s = zm0 * zm0 + zm1 * zm1 + zm2 * zm2 + zm3 * zm3;
    for (int off = 16; off >= 1; off >>= 1) s += __shfl_xor(s, off, 32);
    if (lane == 0) zzW[idx] = s;
}

// preload the per-lane WMMA A fragments (whole K=128) into registers
#define LOAD_AFRAG(afrag, Er, kb)                                  \
    _Pragma("unroll")                                              \
    for (int kk = 0; kk < 32; ++kk) {                              \
        afrag[kk].x = (Er)[kk * 4 + (kb)];                         \
        afrag[kk].y = (Er)[kk * 4 + (kb) + 1];                     \
    }

// ---------------- loss_ip: WMMA emb x proxies^T + streaming LSE -------------
// grid: N/16 blocks x 256 threads. wave w streams proxies [w*1024, w*1024+1024)
__global__ void k_loss_ip(const float* __restrict__ E, const float* __restrict__ Z,
                          const float* __restrict__ xx, const float* __restrict__ zz,
                          const int* __restrict__ items, float* __restrict__ acc) {
    __shared__ float red_m[8][16], red_s[8][16], red_l[8][16];
    const int wave = threadIdx.x >> 5;
    const int lane = threadIdx.x & 31;
    const int r = lane & 15, half = lane >> 4, kb = half * 2;
    const int row0 = blockIdx.x * 16;

    const float* Er = E + (size_t)(row0 + r) * D_;
    v2f afrag[32];
    LOAD_AFRAG(afrag, Er, kb)

    float m[8], s[8], lab[8], xr[8];
    int   labc[8];
#pragma unroll
    for (int v = 0; v < 8; ++v) {
        const int row = row0 + half * 8 + v;
        m[v] = NEG_BIG; s[v] = 0.f; lab[v] = NEG_BIG;
        xr[v] = xx[row]; labc[v] = items[row];
    }
    for (int t = wave * 64; t < wave * 64 + 64; ++t) {
        const int col0 = t * 16;
        v8f d = {0.f, 0.f, 0.f, 0.f, 0.f, 0.f, 0.f, 0.f};
        const float* Zc = Z + (size_t)(col0 + r) * D_;   // B[k,col] = Z[col][k]
#pragma unroll
        for (int kk = 0; kk < 32; ++kk) {
            v2f b;
            b.x = Zc[kk * 4 + kb];
            b.y = Zc[kk * 4 + kb + 1];
            d = wmma_f32(afrag[kk], b, d);
        }
        const float zzc = zz[col0 + r];
#pragma unroll
        for (int v = 0; v < 8; ++v) {
            float l = 2.f * d[v] - xr[v] - zzc;
            if (col0 + r == labc[v]) lab[v] = l;
            if (l > m[v]) { s[v] = s[v] * __expf(m[v] - l) + 1.f; m[v] = l; }
            else           s[v] += __expf(l - m[v]);
        }
    }
    // reduce the 16 column-lanes within each half
    for (int off = 1; off < 16; off <<= 1) {
#pragma unroll
        for (int v = 0; v < 8; ++v) {
            float om = __shfl_xor(m[v], off, 32);
            float os = __shfl_xor(s[v], off, 32);
            float ol = __shfl_xor(lab[v], off, 32);
            float nm = fmaxf(m[v], om);
            s[v] = s[v] * __expf(m[v] - nm) + os * __expf(om - nm);
            m[v] = nm;
            lab[v] = fmaxf(lab[v], ol);
        }
    }
    if (r == 0) {
#pragma unroll
        for (int v = 0; v < 8; ++v) {
            red_m[wave][half * 8 + v] = m[v];
            red_s[wave][half * 8 + v] = s[v];
            red_l[wave][half * 8 + v] = lab[v];
        }
    }
    __syncthreads();
    if (threadIdx.x < 16) {
        const int rr = threadIdx.x;
        float M = NEG_BIG, S = 0.f, L = NEG_BIG;
        for (int w = 0; w < 8; ++w) {
            float om = red_m[w][rr], os = red_s[w][rr];
            float nm = fmaxf(M, om);
            S = S * __expf(M - nm) + os * __expf(om - nm);
            M = nm;
            L = fmaxf(L, red_l[w][rr]);
        }
        atomicAdd(&acc[0], (M + __logf(S)) - L);
    }
}

// ---------------- loss_cp: 64 distinct centers + multiplicity weights -------
// grid: N/16 blocks x 32 threads (1 wave): 4 col tiles cover C=64 categories.
__global__ void k_loss_cp(const float* __restrict__ E, const float* __restrict__ SC,
                          const float* __restrict__ xx, const float* __restrict__ zz_sc,
                          const float* __restrict__ catcnt, const int* __restrict__ cats,
                          float* __restrict__ acc) {
    const int lane = threadIdx.x & 31;
    const int r = lane & 15, half = lane >> 4, kb = half * 2;
    const int row0 = blockIdx.x * 16;
    const float* Er = E + (size_t)(row0 + r) * D_;
    v2f afrag[32];
    LOAD_AFRAG(afrag, Er, kb)

    float m[8], s[8], lab[8], xr[8];
    int   labc[8];
#pragma unroll
    for (int v = 0; v < 8; ++v) {
        const int row = row0 + half * 8 + v;
        m[v] = NEG_BIG; s[v] = 0.f; lab[v] = NEG_BIG;
        xr[v] = xx[row];
        labc[v] = cats[cats[row]];   // label column j=cats[row] uses center cats[j]
    }
    for (int t = 0; t < 4; ++t) {
        v8f d = {0.f, 0.f, 0.f, 0.f, 0.f, 0.f, 0.f, 0.f};
        const int c = t * 16 + r;
        const float* Sc = SC + (size_t)c * D_;
#pragma unroll
        for (int kk = 0; kk < 32; ++kk) {
            v2f b;
            b.x = Sc[kk * 4 + kb];
            b.y = Sc[kk * 4 + kb + 1];
            d = wmma_f32(afrag[kk], b, d);
        }
        const float zzc = zz_sc[c];
        const float cnt = catcnt[c];
#pragma unroll
        for (int v = 0; v < 8; ++v) {
            float l = 2.f * d[v] - xr[v] - zzc;
            if (c == labc[v]) lab[v] = l;
            if (l > m[v]) { s[v] = s[v] * __expf(m[v] - l) + cnt; m[v] = l; }
            else           s[v] += cnt * __expf(l - m[v]);
        }
    }
    for (int off = 1; off < 16; off <<= 1) {
#pragma unroll
        for (int v = 0; v < 8; ++v) {
            float om = __shfl_xor(m[v], off, 32);
            float os = __shfl_xor(s[v], off, 32);
            float ol = __shfl_xor(lab[v], off, 32);
            float nm = fmaxf(m[v], om);
            s[v] = s[v] * __expf(m[v] - nm) + os * __expf(om - nm);
            m[v] = nm;
            lab[v] = fmaxf(lab[v], ol);
        }
    }
    if (r == 0) {
#pragma unroll
        for (int v = 0; v < 8; ++v)
            atomicAdd(&acc[1], (m[v] + __logf(s[v])) - lab[v]);
    }
}

// ---------------- loss_mm: 4 masked heads, logits = +dist^2 -----------------
// grid: A * N/16 blocks x 32 threads; 16 col tiles cover V=256 values.
__global__ void k_loss_mm(const float* __restrict__ E, const float* __restrict__ W2,
                          const float* __restrict__ mxx, const float* __restrict__ zzW,
                          const int* __restrict__ attrs, const float* __restrict__ m_attrs,
                          float* __restrict__ acc) {
    const int a  = blockIdx.x >> 8;
    const int rt = blockIdx.x & 255;
    const int lane = threadIdx.x & 31;
    const int r = lane & 15, half = lane >> 4, kb = half * 2;
    const int row0 = rt * 16;
    const float* Er = E + (size_t)(row0 + r) * D_;
    v2f afrag[32];
    LOAD_AFRAG(afrag, Er, kb)

    float m[8], s[8], lab[8], mx[8], w[8];
    int   labv[8];
#pragma unroll
    for (int v = 0; v < 8; ++v) {
        const int row = row0 + half * 8 + v;
        m[v] = NEG_BIG; s[v] = 0.f; lab[v] = NEG_BIG;
        mx[v]   = mxx[a * N_ + row];
        labv[v] = attrs[a * N_ + row];
        float nf = m_attrs[row] + m_attrs[N_ + row] + m_attrs[2 * N_ + row] + m_attrs[3 * N_ + row];
        w[v] = m_attrs[a * N_ + row] / (nf + 1e-8f);
    }
    for (int t = 0; t < 16; ++t) {
        v8f d = {0.f, 0.f, 0.f, 0.f, 0.f, 0.f, 0.f, 0.f};
        const int c = t * 16 + r;
        const float* Wc = W2 + (size_t)(a * V_ + c) * D_;
#pragma unroll
        for (int kk = 0; kk < 32; ++kk) {
            v2f b;
            b.x = Wc[kk * 4 + kb];
            b.y = Wc[kk * 4 + kb + 1];
            d = wmma_f32(afrag[kk], b, d);
        }
        const float zc = zzW[a * V_ + c];
#pragma unroll
        for (int v = 0; v < 8; ++v) {
            float l = mx[v] - 2.f * d[v] + zc;      // +dist^2 (double negation)
            if (c == labv[v]) lab[v] = l;
            if (l > m[v]) { s[v] = s[v] * __expf(m[v] - l) + 1.f; m[v] = l; }
            else           s[v] += __expf(l - m[v]);
        }
    }
    for (int off = 1; off < 16; off <<= 1) {
#pragma unroll
        for (int v = 0; v < 8; ++v) {
            float om = __shfl_xor(m[v], off, 32);
            float os = __shfl_xor(s[v], off, 32);
            float ol = __shfl_xor(lab[v], off, 32);
            float nm = fmaxf(m[v], om);
            s[v] = s[v] * __expf(m[v] - nm) + os * __expf(om - nm);
            m[v] = nm;
            lab[v] = fmaxf(lab[v], ol);
        }
    }
    if (r == 0) {
#pragma unroll
        for (int v = 0; v < 8; ++v)
            atomicAdd(&acc[2], w[v] * ((m[v] + __logf(s[v])) - lab[v]));
    }
}

// ---------------- finalize ----------------
__global__ void k_finalize(const float* __restrict__ acc, float* __restrict__ out) {
    if (threadIdx.x == 0 && blockIdx.x == 0) {
        out[0] = acc[0] * (1.0f / N_);
        out[1] = acc[1] * (1.0f / N_);
        out[2] = acc[2] * (1.0f / N_);
        out[3] = acc[3] * (1.0f / N_);
    }
}

extern "C" void kernel_launch(void* const* d_in, const int* in_sizes, int n_in,
                              void* d_out, int out_size, void* d_ws, size_t ws_size,
                              hipStream_t stream) {
    const float* images  = (const float*)d_in[0];
    const float* W_model = (const float*)d_in[1];
    const float* proxies = (const float*)d_in[2];
    const float* masks   = (const float*)d_in[3];
    const float* adj     = (const float*)d_in[4];
    const float* m_attrs = (const float*)d_in[5];
    const int*   items   = (const int*)d_in[6];
    const int*   cats    = (const int*)d_in[7];
    const int*   attrs   = (const int*)d_in[8];
    const int*   pcats   = (const int*)d_in[9];

    float* ws     = (float*)d_ws;
    float* E      = ws + O_EMB;
    float* xx     = ws + O_XX;
    float* zzp    = ws + O_ZZP;
    float* sc     = ws + O_SC;
    float* sccnt  = ws + O_SCCNT;
    float* catcnt = ws + O_CATCNT;
    float* acc    = ws + O_ACC;
    float* zzsc   = ws + O_ZZSC;
    float* W2     = ws + O_W2;
    float* zzW    = ws + O_ZZW;
    float* mxx    = ws + O_MXX;
    float* out    = (float*)d_out;

    // zero segment sums / counts / accumulators (contiguous block)
    k_zero<<<(ZERO_LEN + 255) / 256, 256, 0, stream>>>(sc, ZERO_LEN);

    k_gemm_emb<<<N_ / 16, 256, 0, stream>>>(images, W_model, E);
    k_row_stats<<<N_ / 8, 256, 0, stream>>>(E, masks, cats, xx, mxx, catcnt, acc);
    k_proxy_stats<<<P_ / 8, 256, 0, stream>>>(proxies, pcats, zzp, sc, sccnt);
    k_centers<<<C_ / 8, 256, 0, stream>>>(sc, sccnt, zzsc);
    k_mask_adj<<<(A_ * V_) / 8, 256, 0, stream>>>(adj, masks, W2, zzW);

    k_loss_ip<<<N_ / 16, 256, 0, stream>>>(E, proxies, xx, zzp, items, acc);
    k_loss_cp<<<N_ / 16, 32, 0, stream>>>(E, sc, xx, zzsc, catcnt, cats, acc);
    k_loss_mm<<<A_ * (N_ / 16), 32, 0, stream>>>(E, W2, mxx, zzW, attrs, m_attrs, acc);

    k_finalize<<<1, 32, 0, stream>>>(acc, out);
}